// SAGEAggregator_7602092114107
// MI455X (gfx1250) — compile-verified
//
#include <hip/hip_runtime.h>

typedef __attribute__((ext_vector_type(2))) float v2f;
typedef __attribute__((ext_vector_type(4))) float v4f;
typedef __attribute__((ext_vector_type(8))) float v8f;

#define NNODES 50000
#define KNEIGH 32
#define DIN    128
#define DOUT   128
#define D2     256                   // contraction length of fused GEMM: [x | agg]
#define TILE_M 16
#define NTILES (NNODES / TILE_M)     // 3125 (exact)
#define APAD   260                   // row pitch of A tile in LDS (floats), avoids bank conflicts
#define NWAVES 8
#define KSTEPS (D2 / 4)              // 64 wmma_f32_16x16x4 steps per output tile

__global__ __launch_bounds__(256) void sage_fused_kernel(
    const float* __restrict__ x,
    const float* __restrict__ neigh,
    const float* __restrict__ Wl,
    const float* __restrict__ Wr,
    float* __restrict__ out)
{
    // A tile: 16 nodes x 256 (x concat mean-agg), padded rows
    __shared__ float A_lds[TILE_M * APAD];                    // 16.6 KB
    // Pre-swizzled B fragments: [wave][kstep][lane][2]  -> conflict-free b64 reads
    __shared__ float Bfrag[NWAVES * KSTEPS * 32 * 2];         // 128 KB

    const int tid  = threadIdx.x;
    const int wave = tid >> 5;     // wave32
    const int lane = tid & 31;

    // ---- one-time per block: build swizzled combined weight [W_l ; W_r] in LDS ----
    // element i = ((w*64 + ks)*32 + ln)*2 + j  holds  B[K][N] with
    //   K = 4*ks + 2*(ln>>4) + j   (ISA f32 B-matrix 4x16 lane layout)
    //   N = w*16 + (ln & 15)
    for (int i = tid; i < NWAVES * KSTEPS * 32 * 2; i += 256) {
        const int j  = i & 1;
        const int ln = (i >> 1) & 31;
        const int ks = (i >> 6) & 63;
        const int w  = i >> 12;
        const int kk = 4 * ks + 2 * (ln >> 4) + j;   // 0..255
        const int oo = w * 16 + (ln & 15);           // 0..127
        Bfrag[i] = (kk < DIN) ? Wl[oo * DIN + kk] : Wr[oo * DIN + (kk - DIN)];
    }
    __syncthreads();

    const int qd   = tid & 31;     // v4f column within a 128-wide row
    const int slot = tid >> 5;     // 0..7 -> node sub-slot

    for (int tile = blockIdx.x; tile < NTILES; tile += gridDim.x) {
        const int node0 = tile * TILE_M;

        // ---- streaming aggregation (the bandwidth-bound part):
        //      A[m][0..127] = x[node0+m], A[m][128..255] = mean_k neigh[node0+m][k]
        //      neigh_x is read exactly once -> non-temporal (keep L2 for weights)
        #pragma unroll
        for (int s = 0; s < 2; ++s) {
            const int m = slot + 8 * s;
            const v4f* nptr =
                (const v4f*)(neigh + (size_t)(node0 + m) * (KNEIGH * DIN)
                             + (size_t)(qd * 4));
            v4f acc = {0.f, 0.f, 0.f, 0.f};
            #pragma unroll 8
            for (int k = 0; k < KNEIGH; ++k) {
                acc += __builtin_nontemporal_load(nptr + k * (DIN / 4));
            }
            acc *= (1.0f / (float)KNEIGH);
            const v4f xx = __builtin_nontemporal_load(
                (const v4f*)(x + (size_t)(node0 + m) * DIN + qd * 4));
            *(v4f*)(&A_lds[m * APAD + qd * 4])       = xx;
            *(v4f*)(&A_lds[m * APAD + DIN + qd * 4]) = acc;
        }
        __syncthreads();

        // ---- GEMM: each wave produces a 16(node) x 16(out-feature) tile ----
        v8f c = {};
        const int am   = lane & 15;   // A row (M) / D column (N)
        const int half = lane >> 4;   // selects K pair per ISA layout
        const float* arow = &A_lds[am * APAD + 2 * half];
        const v2f*   bptr = (const v2f*)&Bfrag[wave * (KSTEPS * 64)];
        #pragma unroll 4
        for (int ks = 0; ks < KSTEPS; ++ks) {
            const v2f a = *(const v2f*)(arow + 4 * ks);   // A[am][4*ks + 2*half + {0,1}]
            const v2f b = bptr[ks * 32 + lane];           // swizzled B fragment
            c = __builtin_amdgcn_wmma_f32_16x16x4_f32(
                    /*neg_a=*/false, a, /*neg_b=*/false, b,
                    /*c_mod=*/(short)0, c, /*reuse_a=*/false, /*reuse_b=*/false);
        }

        // ---- store D: VGPR r -> out[node0 + r + 8*half][wave*16 + am] ----
        // One base pointer + constant 512 B strides -> immediate ioffsets.
        float* obase = out + (size_t)(node0 + 8 * half) * DOUT + (wave * 16 + am);
        #pragma unroll
        for (int r = 0; r < 8; ++r) {
            __builtin_nontemporal_store(c[r], obase + r * DOUT);
        }
        __syncthreads();   // protect A_lds before next tile's aggregation
    }
}

extern "C" void kernel_launch(void* const* d_in, const int* in_sizes, int n_in,
                              void* d_out, int out_size, void* d_ws, size_t ws_size,
                              hipStream_t stream) {
    const float* x     = (const float*)d_in[0];  // [50000,128]
    const float* neigh = (const float*)d_in[1];  // [50000,32,128]
    const float* Wl    = (const float*)d_in[2];  // [128,128]
    const float* Wr    = (const float*)d_in[3];  // [128,128]
    float* out = (float*)d_out;                  // [50000,128]

    // Grid-stride over 3125 node tiles; 1024 blocks amortizes the per-block
    // 128 KB weight staging ~3x while keeping every WGP saturated.
    sage_fused_kernel<<<dim3(1024), dim3(256), 0, stream>>>(x, neigh, Wl, Wr, out);
}